// Lorentz_9552007266545
// MI455X (gfx1250) — compile-verified
//
#include <hip/hip_runtime.h>

// CDNA5 / gfx1250 Lorentz boost kernel.
// out[b,c,i] = sum_j Bmat[c][i][j] * T[b,c,j],  T: (8192,1024,4) f32, Bo: (1024,3) f32
//
// Memory-bound (~1 FLOP/byte): single pass over 256 MiB, floor ~11.5us @ 23.3 TB/s.
// Compute core: V_WMMA_F32_16X16X4_F32 with a block-diagonal K split so that
// A fragments are contiguous float2 global loads (no shuffles) and every element
// of the 16x16 D tile is a useful output (16 batches x 4 channels x 4 components).
// NCHAN specialization makes the row stride a compile-time constant so the 8
// D-tile stores share one address VGPR pair with immediate offsets.

typedef __attribute__((ext_vector_type(2))) float v2f;
typedef __attribute__((ext_vector_type(8))) float v8f;

#define EPSF 1e-7f

__device__ __forceinline__ float boost_elem(int row, int col, float g,
                                            float b0, float b1, float b2,
                                            float n0, float n1, float n2) {
  // Bmat[row][col] of B = I - (g*mag)*nK + (g-1)*nK^2
  // B[0][0]=g ; B[0][j+1]=B[j+1][0]=-g*Bo_j ; B[i+1][j+1]=delta_ij+(g-1)*n_i*n_j
  float bc = (col == 1) ? b0 : (col == 2) ? b1 : b2;
  float br = (row == 1) ? b0 : (row == 2) ? b1 : b2;
  float nc = (col == 1) ? n0 : (col == 2) ? n1 : n2;
  float nr = (row == 1) ? n0 : (row == 2) ? n1 : n2;
  float v = ((row == col) ? 1.0f : 0.0f) + (g - 1.0f) * nr * nc;
  v = (col == 0) ? (-g * br) : v;
  v = (row == 0) ? (-g * bc) : v;
  v = (row == 0 && col == 0) ? g : v;
  return v;
}

// NCHAN == 0 -> runtime nchan; NCHAN > 0 -> compile-time stride (fast path).
template <int NCHAN>
__global__ __launch_bounds__(256) void lorentz_boost_wmma(
    const float* __restrict__ T, const float* __restrict__ Bo,
    float* __restrict__ out, int batch, int nchan_rt) {
  const int nchan = (NCHAN > 0) ? NCHAN : nchan_rt;
  const long long rowStride = (long long)nchan * 4;  // floats per batch row

  const int lane = threadIdx.x & 31;
  const int wave = threadIdx.x >> 5;
  const int cg   = blockIdx.x * 8 + wave;   // channel group of 4 channels
  if (cg * 4 >= nchan) return;              // uniform per wave
  const int c0 = cg * 4;

  const int n  = lane & 15;                 // D column / B column this lane owns
  const int h  = lane >> 4;                 // half-wave: K pair {0,1} or {2,3}
  const int k0 = 2 * h;

  // ---- per-wave B fragment construction (runs once; ~nothing vs. 256 MiB stream)
  const int c = c0 + (n >> 2);              // channel owned by this lane's column
  const int r = n & 3;                      // row of Bmat owned by this column
  float bo0 = Bo[c * 3 + 0], bo1 = Bo[c * 3 + 1], bo2 = Bo[c * 3 + 2];
  float mag = sqrtf(bo0 * bo0 + bo1 * bo1 + bo2 * bo2);
  mag = fminf(fmaxf(mag, EPSF), 1.0f - EPSF);
  float inv = 1.0f / mag;
  float n0 = bo0 * inv, n1 = bo1 * inv, n2 = bo2 * inv;
  float g = 1.0f / sqrtf(1.0f - mag * mag);

  float e0 = boost_elem(r, k0 + 0, g, bo0, bo1, bo2, n0, n1, n2);
  float e1 = boost_elem(r, k0 + 1, g, bo0, bo1, bo2, n0, n1, n2);

  const int mygrp = n >> 2;
  v2f bsel[4];
#pragma unroll
  for (int gi = 0; gi < 4; ++gi) {
    bsel[gi].x = (mygrp == gi) ? e0 : 0.0f;  // B_g[k, n] block-diagonal mask
    bsel[gi].y = (mygrp == gi) ? e1 : 0.0f;
  }

  // ---- stream batch tiles of 16 rows with pure pointer-increment addressing
  const int tilesTotal = batch >> 4;
  const long long step = (long long)gridDim.y * 16 * rowStride;  // floats/iter

  // A fragment for this lane: T[b0 + n, c0 + g, k0..k0+1] -> contiguous float2
  const float* aptr = T + (long long)(blockIdx.y * 16 + n) * rowStride +
                      (c0 << 2) + k0;
  // D[m, col] -> out[(b0+m)*rowStride + c0*4 + col]; VGPR v holds row v + 8*h
  float* optr = out + (long long)(blockIdx.y * 16 + 8 * h) * rowStride +
                (c0 << 2) + n;

  for (int t = blockIdx.y; t < tilesTotal; t += (int)gridDim.y) {
    // Speculative SYS-scope prefetch of the next tile (silently dropped if OOB).
    __builtin_prefetch(aptr + step, 0, 0);  // global_prefetch_b8

    v2f a0 = *(const v2f*)(aptr + 0);
    v2f a1 = *(const v2f*)(aptr + 4);
    v2f a2 = *(const v2f*)(aptr + 8);
    v2f a3 = *(const v2f*)(aptr + 12);

    v8f acc = {};
    acc = __builtin_amdgcn_wmma_f32_16x16x4_f32(false, a0, false, bsel[0],
                                                (short)0, acc, false, false);
    acc = __builtin_amdgcn_wmma_f32_16x16x4_f32(false, a1, false, bsel[1],
                                                (short)0, acc, false, false);
    acc = __builtin_amdgcn_wmma_f32_16x16x4_f32(false, a2, false, bsel[2],
                                                (short)0, acc, false, false);
    acc = __builtin_amdgcn_wmma_f32_16x16x4_f32(false, a3, false, bsel[3],
                                                (short)0, acc, false, false);

#pragma unroll
    for (int v = 0; v < 8; ++v) {
      // With NCHAN compile-time, v*rowStride*4 <= 112 KiB folds into the
      // 24-bit immediate offset: 8 stores share one address VGPR pair.
      optr[(long long)v * rowStride] = acc[v];
    }

    aptr += step;
    optr += step;
  }
}

extern "C" void kernel_launch(void* const* d_in, const int* in_sizes, int n_in,
                              void* d_out, int out_size, void* d_ws, size_t ws_size,
                              hipStream_t stream) {
  const float* T  = (const float*)d_in[0];
  const float* Bo = (const float*)d_in[1];
  float* out = (float*)d_out;

  const int nchan = in_sizes[1] / 3;               // 1024
  const int batch = in_sizes[0] / (nchan * 4);     // 8192

  const int groups = nchan / 4;                    // channel groups of 4
  dim3 block(256);                                 // 8 waves: adjacent groups
  dim3 grid(groups / 8, 64);                       // 8 batch tiles per wave

  if (nchan == 1024) {
    lorentz_boost_wmma<1024><<<grid, block, 0, stream>>>(T, Bo, out, batch, nchan);
  } else {
    lorentz_boost_wmma<0><<<grid, block, 0, stream>>>(T, Bo, out, batch, nchan);
  }
}